// GroupedLinear_73658689126883
// MI455X (gfx1250) — compile-verified
//
#include <hip/hip_runtime.h>

// Grouped (block-diagonal) linear + channel shuffle for MI455X (gfx1250).
// x:[16,4096,512] f32, W:[8,64,64] f32, b:[8,64] f32 -> out:[16,4096,512] f32
// out[t, h*8+g] = sum_i x[t, g*64+i] * W[g,h,i] + b[g,h]
//
// Memory-bound (≈268 MB @ 23.3 TB/s ≈ 11.5us). Uses V_WMMA_F32_16X16X4_F32,
// weight-stationary B fragments, LDS-staged coalesced shuffled stores.

typedef float v2f __attribute__((ext_vector_type(2)));
typedef float v4f __attribute__((ext_vector_type(4)));
typedef float v8f __attribute__((ext_vector_type(8)));

#define GRP 8
#define IPG 64   // inputs per group
#define HPG 64   // hidden per group
#define ROW 512  // GRP * IPG == GRP * HPG

__global__ __launch_bounds__(256) void grouped_linear_wmma(
    const float* __restrict__ x, const float* __restrict__ W,
    const float* __restrict__ bias, float* __restrict__ out, int ntiles) {
  // 16 tokens x 512 cols staging buffer for the channel shuffle (32 KB of LDS)
  __shared__ float lds[16 * ROW];

  const int lane = threadIdx.x & 31;
  const int g    = threadIdx.x >> 5;  // wave id == group id (8 waves / block)
  const int half = lane >> 4;         // 0: lanes 0-15, 1: lanes 16-31
  const int l16  = lane & 15;

  // ---- Weight-stationary: preload full 64x64 W_g as WMMA B fragments ----
  // B is 4x16 (KxN) per step. VGPR v, lane half h: element (K = v + 2*h, N = l16).
  // B[k_step][v] at lane = W[g, n*16+l16, 4*k_step + 2*half + v]
  v2f bfrag[4][16];
  const float* Wg = W + g * HPG * IPG;
#pragma unroll
  for (int n = 0; n < 4; ++n) {
    const float* wr = Wg + (n * 16 + l16) * IPG + half * 2;
#pragma unroll
    for (int k = 0; k < 16; ++k) bfrag[n][k] = *(const v2f*)(wr + 4 * k);
  }

  // Bias: D element (v, lane) has h = n*16 + l16 (same for all v) -> one
  // scalar per lane per n-tile; seed the accumulator with it (free bias add).
  float bias_f[4];
#pragma unroll
  for (int n = 0; n < 4; ++n) bias_f[n] = bias[g * HPG + n * 16 + l16];

  for (int tile = blockIdx.x; tile < ntiles; tile += (int)gridDim.x) {
    const long tok0 = (long)tile * 16;

    // ---- A fragments: 16 tokens x 64 inputs of this group ----
    // A is 16x4 (MxK): VGPR v, lane half h: element (M = l16, K = v + 2*h).
    const float* xr = x + (tok0 + l16) * ROW + g * IPG + half * 2;
    v2f afrag[16];
#pragma unroll
    for (int k = 0; k < 16; ++k) afrag[k] = *(const v2f*)(xr + 4 * k);

    // Prefetch next tile's slice for this wave (global_prefetch_b8)
    if (tile + (int)gridDim.x < ntiles)
      __builtin_prefetch(xr + (long)gridDim.x * 16 * ROW, 0, 3);

    // ---- 4 N-tiles x 16 K-steps of fp32 WMMA ----
#pragma unroll
    for (int n = 0; n < 4; ++n) {
      v8f acc;
#pragma unroll
      for (int i = 0; i < 8; ++i) acc[i] = bias_f[n];
#pragma unroll
      for (int k = 0; k < 16; ++k)
        acc = __builtin_amdgcn_wmma_f32_16x16x4_f32(
            false, afrag[k], false, bfrag[n][k], (short)0, acc, false, false);

      // Scatter D into LDS with the channel shuffle applied:
      // D element (v, lane): token m = v + 8*half, h = n*16 + l16.
      // shuffled column = h*GRP + g
#pragma unroll
      for (int v = 0; v < 8; ++v)
        lds[(v + half * 8) * ROW + (n * 16 + l16) * GRP + g] = acc[v];
    }

    __syncthreads();

    // ---- Coalesced b128 streaming store of the staged 16x512 tile ----
    {
      float* orow = out + tok0 * ROW;
#pragma unroll
      for (int it = 0; it < 8; ++it) {
        int idx = (it * 256 + (int)threadIdx.x) * 4;  // float4 index
        *(v4f*)(orow + idx) = *(const v4f*)(lds + idx);
      }
    }

    __syncthreads();  // LDS reuse guard for next tile
  }
}

extern "C" void kernel_launch(void* const* d_in, const int* in_sizes, int n_in,
                              void* d_out, int out_size, void* d_ws, size_t ws_size,
                              hipStream_t stream) {
  const float* x    = (const float*)d_in[0];
  const float* W    = (const float*)d_in[1];
  const float* bias = (const float*)d_in[2];
  float* out        = (float*)d_out;

  const int ntok   = in_sizes[0] / ROW;  // 65536 tokens
  const int ntiles = ntok / 16;          // 4096 16-token tiles

  int blocks = ntiles < 512 ? ntiles : 512;  // ~8 tiles per block: amortize W preload
  grouped_linear_wmma<<<dim3(blocks), dim3(256), 0, stream>>>(x, W, bias, out, ntiles);
}